// LatentTreeGCN_59820304499129
// MI455X (gfx1250) — compile-verified
//
#include <hip/hip_runtime.h>
#include <hip/hip_bf16.h>
#include <math.h>

// ---------------------------------------------------------------------------
// LatentTreeGCN for MI455X (gfx1250, wave32, WMMA)
//   B=8, N=128, H=256
//   GEMMs -> V_WMMA_F32_16X16X4_F32, 64x16 tile per wave (4 accumulators
//            sharing one B fragment -> 1.5 loads per WMMA)
//   MTT marginals -> closed form via 128x128 Gauss-Jordan inverse in LDS,
//                    with output concat fused into the same kernel.
//   einsum('bnm,bnh->bnh') == X * rowsum(P), so P is never materialized.
// ---------------------------------------------------------------------------

typedef __attribute__((ext_vector_type(2))) float v2f;
typedef __attribute__((ext_vector_type(8))) float v8f;

#define BSZ   8
#define NSEQ  128
#define HD    256
#define ROWS  (BSZ * NSEQ)          // 1024
#define LDW   129                   // padded LDS row stride (bank-conflict free)

// workspace layout (float elements)
#define WS_AC      0                              // 4 matrices [ROWS][HD] : a_p, c_p, a_h, c_h
#define WS_SCORES  (4 * ROWS * HD)                // 2*8 matrices [128][128]

// ---------------------------------------------------------------------------
// Kernel 1: a/c = X @ W1_half  via f32 WMMA.
//   grid = (2, 16, 4): x -> 512-row M block (8 waves x 64 rows),
//                      y -> 16-col N tile,
//                      z -> {p,h} x {W1[:H], W1[H:]}
//   Each wave: 64x16 tile = 4 stacked 16x16 WMMA tiles sharing the B fragment.
// A fragment (16x4 f32): lanes 0-15 = rows, lane-half selects K pair (VGPR0/1).
// B fragment (4x16 f32): lanes 0-15 = cols, lane-half selects K pair.
// C/D (16x16 f32): lane = col, vreg v = row v (+8 for upper lane half).
// ---------------------------------------------------------------------------
__global__ __launch_bounds__(256)
void gemm_ac_wmma(const float* __restrict__ Xp, const float* __restrict__ Xh,
                  const float* __restrict__ W1, float* __restrict__ ws)
{
    const int z    = blockIdx.z;        // 0..3
    const int inp  = z >> 1;
    const int hw   = z & 1;
    const float* X = inp ? Xh : Xp;
    const float* W = W1 + (size_t)hw * HD * HD;   // W1 is [2H, H] row-major [in,out]
    float* Out     = ws + WS_AC + (size_t)z * ROWS * HD;

    const int wave = threadIdx.x >> 5;
    const int lane = threadIdx.x & 31;
    const int lh   = lane >> 4;         // 0/1 : K-pair select
    const int l16  = lane & 15;

    const int m0 = blockIdx.x * 512 + wave * 64;  // 4 sub-tiles of 16 rows
    const int n0 = blockIdx.y * 16;

    const float* Ar0 = X + (size_t)(m0 +  0 + l16) * HD;
    const float* Ar1 = X + (size_t)(m0 + 16 + l16) * HD;
    const float* Ar2 = X + (size_t)(m0 + 32 + l16) * HD;
    const float* Ar3 = X + (size_t)(m0 + 48 + l16) * HD;

    v8f acc0 = {0.f,0.f,0.f,0.f,0.f,0.f,0.f,0.f};
    v8f acc1 = acc0, acc2 = acc0, acc3 = acc0;

    for (int k = 0; k < HD; k += 4) {
        const int ka = k + 2 * lh;
        v2f bfrag;
        bfrag.x = W[(size_t)ka * HD + n0 + l16];
        bfrag.y = W[(size_t)(ka + 1) * HD + n0 + l16];
        v2f a0 = *(const v2f*)(Ar0 + ka);      // 8B aligned (ka even)
        v2f a1 = *(const v2f*)(Ar1 + ka);
        v2f a2 = *(const v2f*)(Ar2 + ka);
        v2f a3 = *(const v2f*)(Ar3 + ka);
        // 8 args: (neg_a, A, neg_b, B, c_mod, C, reuse_a, reuse_b)
        acc0 = __builtin_amdgcn_wmma_f32_16x16x4_f32(false, a0, false, bfrag,
                                                     (short)0, acc0, false, false);
        acc1 = __builtin_amdgcn_wmma_f32_16x16x4_f32(false, a1, false, bfrag,
                                                     (short)0, acc1, false, false);
        acc2 = __builtin_amdgcn_wmma_f32_16x16x4_f32(false, a2, false, bfrag,
                                                     (short)0, acc2, false, false);
        acc3 = __builtin_amdgcn_wmma_f32_16x16x4_f32(false, a3, false, bfrag,
                                                     (short)0, acc3, false, false);
    }

    float* Ob = Out + (size_t)(m0 + 8 * lh) * HD + n0 + l16;
#pragma unroll
    for (int v = 0; v < 8; ++v) Ob[(size_t)(v +  0) * HD] = acc0[v];
#pragma unroll
    for (int v = 0; v < 8; ++v) Ob[(size_t)(v + 16) * HD] = acc1[v];
#pragma unroll
    for (int v = 0; v < 8; ++v) Ob[(size_t)(v + 32) * HD] = acc2[v];
#pragma unroll
    for (int v = 0; v < 8; ++v) Ob[(size_t)(v + 48) * HD] = acc3[v];
}

// ---------------------------------------------------------------------------
// Kernel 2: scores[b,i,j] = sum_d tanh(a[b,i,d] + c[b,j,d] + b1[d]) * W2[d] + b2
//   grid = (128 i, 8 b, 2 input), 256 threads = 8 waves; one wave per j-stripe.
// ---------------------------------------------------------------------------
__global__ __launch_bounds__(256)
void arc_scores_kernel(const float* __restrict__ b1, const float* __restrict__ W2,
                       const float* __restrict__ b2, float* __restrict__ ws)
{
    __shared__ float arow[HD];
    __shared__ float w2s[HD];

    const int i   = blockIdx.x;
    const int b   = blockIdx.y;
    const int inp = blockIdx.z;
    const float* A = ws + WS_AC + ((size_t)inp * 2 + 0) * ROWS * HD;  // head side
    const float* C = ws + WS_AC + ((size_t)inp * 2 + 1) * ROWS * HD;  // dep side
    const int tid = threadIdx.x;
    const int row = b * NSEQ + i;

    arow[tid] = A[(size_t)row * HD + tid] + b1[tid];
    w2s[tid]  = W2[tid];
    __syncthreads();

    const int wave = tid >> 5, lane = tid & 31;
    const float bias2 = b2[0];
    float* sout = ws + WS_SCORES + ((size_t)(inp * BSZ + b) * NSEQ + i) * NSEQ;

    for (int j = wave; j < NSEQ; j += 8) {
        const float* crow = C + (size_t)(b * NSEQ + j) * HD;
        float s = 0.f;
#pragma unroll
        for (int t = 0; t < 8; ++t) {
            const int d = lane + 32 * t;
            s += tanhf(arow[d] + crow[d]) * w2s[d];
        }
        for (int off = 16; off > 0; off >>= 1)
            s += __shfl_xor(s, off, 32);
        if (lane == 0) sout[j] = s + bias2;
    }
}

// ---------------------------------------------------------------------------
// Kernel 3: matrix-tree marginal row sums + fused output assembly.
//   One block per 128x128 score matrix (16 total). LDS holds L (padded 128x129),
//   inverted in place with Gauss-Jordan; then
//     rowsum[h] = sum_m P[h,m],
//     P[h,m] = A[h,m]*(Linv[m,m]*[m>=1] - Linv[m,h]*[h>=1])  (h != m)
//     P[m,m] = r[m]*Linv[m,0]
//   The max-shift cancels exactly in P (A scales by e^-c, Linv by e^+c).
//   Tail: this block owns rows b*128..b*128+127 of input `inp`, so it writes
//   out[inp][row] = concat(X[row], X[row]*rowsum[row]) directly (rowsum in LDS).
// ---------------------------------------------------------------------------
__global__ __launch_bounds__(256)
void mtt_rowsum_out_kernel(const float* __restrict__ Xp, const float* __restrict__ Xh,
                           float* __restrict__ ws, float* __restrict__ out)
{
    extern __shared__ float smem[];
    float* Wm  = smem;                 // 128 * 129
    float* red = smem + 128 * LDW;     // 256

    const int mat = blockIdx.x;        // 0..15  (inp*8 + b)
    const int inp = mat >> 3;
    const int b   = mat & 7;
    const float* S = ws + WS_SCORES + (size_t)mat * NSEQ * NSEQ;
    const int tid = threadIdx.x;

    // ---- max over all scores (numerical shift) ----
    float pm = -1e30f;
    for (int idx = tid; idx < NSEQ * NSEQ; idx += 256) pm = fmaxf(pm, S[idx]);
    red[tid] = pm;
    __syncthreads();
    for (int s2 = 128; s2 > 0; s2 >>= 1) {
        if (tid < s2) red[tid] = fmaxf(red[tid], red[tid + s2]);
        __syncthreads();
    }
    const float cmax = red[0];
    __syncthreads();

    // ---- exp weights into LDS (diagonal entry holds r[m]) ----
    for (int idx = tid; idx < NSEQ * NSEQ; idx += 256) {
        const int i2 = idx >> 7, j2 = idx & 127;
        Wm[i2 * LDW + j2] = expf(S[idx] - cmax);
    }
    __syncthreads();

    // ---- column sums D[m] = sum_{h != m} A[h,m] ----
    if (tid < NSEQ) {
        const int m = tid;
        float d = 0.f;
        for (int i2 = 0; i2 < NSEQ; ++i2)
            if (i2 != m) d += Wm[i2 * LDW + m];
        red[m] = d;
    }
    __syncthreads();

    // ---- build Laplacian: L[i,j] = -A (i!=j), diag = D, row0 = r ----
    if (tid < NSEQ) {
        const int m = tid;
        const float rm = Wm[m * LDW + m];
        for (int i2 = 1; i2 < NSEQ; ++i2) {
            const float v = (i2 == m) ? red[m] : -Wm[i2 * LDW + m];
            Wm[i2 * LDW + m] = v;
        }
        Wm[m] = rm;   // row 0
    }
    __syncthreads();

    // ---- in-place Gauss-Jordan inversion (no pivoting; L is well conditioned) ----
    const int i     = tid & 127;
    const int halfc = tid >> 7;
    const int j0 = halfc * 64, j1 = j0 + 64;

    for (int k = 0; k < NSEQ; ++k) {
        const float p = Wm[k * LDW + k];
        __syncthreads();
        const float invp = 1.0f / p;
        if (i == k) {
            for (int j = j0; j < j1; ++j)
                Wm[k * LDW + j] = (j == k) ? invp : Wm[k * LDW + j] * invp;
        }
        __syncthreads();
        const float f = (i == k) ? 0.f : Wm[i * LDW + k];
        __syncthreads();
        if (i != k) {
            for (int j = j0; j < j1; ++j) {
                const float mkj = Wm[k * LDW + j];
                const float cur = (j == k) ? 0.f : Wm[i * LDW + j];
                Wm[i * LDW + j] = cur - f * mkj;
            }
        }
        __syncthreads();
    }
    // Wm now holds Linv

    // ---- rowsum[h] = sum_m P[h,m] (two column halves per row) ----
    const int h = i;
    const float rh = expf(S[h * NSEQ + h] - cmax);
    float accum = 0.f;
    for (int m = j0; m < j1; ++m) {
        if (m == h) {
            accum += rh * Wm[h * LDW + 0];
        } else {
            const float a = expf(S[h * NSEQ + m] - cmax);
            const float term = ((m >= 1) ? Wm[m * LDW + m] : 0.f)
                             - ((h >= 1) ? Wm[m * LDW + h] : 0.f);
            accum += a * term;
        }
    }
    red[tid] = accum;
    __syncthreads();
    if (halfc == 0) red[h] = red[h] + red[h + 128];   // final rowsum in red[0..127]
    __syncthreads();

    // ---- fused output: out[inp][b*128+n] = concat(X, X * rowsum[n]) ----
    const float* X = inp ? Xh : Xp;
    const float* Xb = X + (size_t)b * NSEQ * HD;
    float* Ob = out + (size_t)inp * ROWS * 2 * HD + (size_t)b * NSEQ * 2 * HD;
    for (int idx = tid; idx < NSEQ * HD; idx += 256) {
        const int n = idx >> 8;          // 0..127
        const int d = idx & 255;
        const float x = Xb[idx];
        const float r = red[n];
        float* o = Ob + (size_t)n * 2 * HD;
        o[d]      = x;
        o[HD + d] = x * r;
    }
}

// ---------------------------------------------------------------------------
extern "C" void kernel_launch(void* const* d_in, const int* in_sizes, int n_in,
                              void* d_out, int out_size, void* d_ws, size_t ws_size,
                              hipStream_t stream) {
    const float* Xp = (const float*)d_in[0];
    const float* Xh = (const float*)d_in[1];
    const float* W1 = (const float*)d_in[2];
    const float* b1 = (const float*)d_in[3];
    const float* W2 = (const float*)d_in[4];
    const float* b2 = (const float*)d_in[5];
    float* ws  = (float*)d_ws;
    float* out = (float*)d_out;

    dim3 g1(2, 16, 4);
    gemm_ac_wmma<<<g1, 256, 0, stream>>>(Xp, Xh, W1, ws);

    dim3 g2(NSEQ, BSZ, 2);
    arc_scores_kernel<<<g2, 256, 0, stream>>>(b1, W2, b2, ws);

    const size_t shmem = (size_t)(128 * LDW + 256) * sizeof(float);  // ~66 KB (gfx1250: 320 KB/WGP)
    mtt_rowsum_out_kernel<<<16, 256, shmem, stream>>>(Xp, Xh, ws, out);
}